// CommonModel_60481729462377
// MI455X (gfx1250) — compile-verified
//
#include <hip/hip_runtime.h>
#include <hip/hip_bf16.h>

typedef __attribute__((ext_vector_type(2))) float v2f;
typedef __attribute__((ext_vector_type(8))) float v8f;

// ---------------- utility fills ----------------
__global__ void fill_f32(float* p, float v, long n) {
  long i = (long)blockIdx.x * blockDim.x + threadIdx.x;
  if (i < n) p[i] = v;
}
__global__ void fill_u32(unsigned* p, unsigned v, long n) {
  long i = (long)blockIdx.x * blockDim.x + threadIdx.x;
  if (i < n) p[i] = v;
}

// ---------------- SAGE mean-aggregation (scatter) ----------------
// thread t -> edge e = t>>kshift, feature k = t & (K-1); coalesced atomics.
__global__ void sage_agg(const int* __restrict__ src, const int* __restrict__ dst,
                         const float* __restrict__ x, float* acc, float* cnt,
                         long E, int kshift) {
  long t = (long)blockIdx.x * blockDim.x + threadIdx.x;
  long e = t >> kshift;
  if (e >= E) return;
  int K = 1 << kshift;
  int k = (int)(t & (K - 1));
  int s = src[e], d = dst[e];
  atomicAdd(&acc[(long)d * K + k], x[(long)s * K + k]);
  if (k == 0) atomicAdd(&cnt[d], 1.0f);
}

// ---------------- fused dual-GEMM + bias + relu via fp32 WMMA ----------------
// out[N,64] = act( (A1 * (cnt?1/max(cnt,1):1)) @ WL  +  (HAS_A2 ? A2 @ WR : 0) + bias )
// One wave -> 16 rows x 64 cols (4 D-tiles), K in steps of 4 (v_wmma_f32_16x16x4_f32).
// K and HAS_A2 are compile-time so all register arrays have static indices (no movrel).
template <int K, bool HAS_A2>
__global__ __launch_bounds__(256) void linear_wmma(
    const float* __restrict__ A1, const float* __restrict__ cnt,
    const float* __restrict__ A2,
    const float* __restrict__ WL, const float* __restrict__ WR,
    const float* __restrict__ bias, float* __restrict__ out,
    int N, int relu) {
  constexpr int NK = K / 4;            // k-steps of 4 (8 or 16)
  constexpr int NW = K * 64;
  __shared__ float wlds[(HAS_A2 ? 2 : 1) * NW];
  for (int i = threadIdx.x; i < NW; i += blockDim.x) wlds[i] = WL[i];
  if (HAS_A2)
    for (int i = threadIdx.x; i < NW; i += blockDim.x) wlds[NW + i] = WR[i];
  __syncthreads();

  int lane = threadIdx.x & 31;
  int wave = threadIdx.x >> 5;
  int row_base = (blockIdx.x * 8 + wave) * 16;
  if (row_base >= N) return;           // wave-uniform: EXEC stays all-ones

  int half = lane >> 4;                // 0: K={0,1} of each 4-step, 1: K={2,3}
  int lr   = lane & 15;
  int m    = row_base + lr;            // A row owned by this lane

  float inv = 1.0f;
  if (cnt) inv = 1.0f / fmaxf(cnt[m], 1.0f);

  v2f aM[NK];
  v2f aX[HAS_A2 ? NK : 1];
  {
    const float* a1row = A1 + (long)m * K + half * 2;
#pragma unroll
    for (int kk = 0; kk < NK; ++kk) {
      v2f a = *(const v2f*)(a1row + kk * 4);
      aM[kk] = a * inv;                // fold mean into A
    }
    if (HAS_A2) {
      const float* a2row = A2 + (long)m * K + half * 2;
#pragma unroll
      for (int kk = 0; kk < NK; ++kk) aX[kk] = *(const v2f*)(a2row + kk * 4);
    }
  }

#pragma unroll
  for (int t = 0; t < 4; ++t) {
    int col = t * 16 + lr;
    v8f c = {0.f, 0.f, 0.f, 0.f, 0.f, 0.f, 0.f, 0.f};
#pragma unroll
    for (int kk = 0; kk < NK; ++kk) {
      int k0 = kk * 4 + half * 2;
      v2f bL = {wlds[k0 * 64 + col], wlds[(k0 + 1) * 64 + col]};
      c = __builtin_amdgcn_wmma_f32_16x16x4_f32(false, aM[kk], false, bL,
                                                (short)0, c, false, false);
      if (HAS_A2) {
        v2f bR = {wlds[NW + k0 * 64 + col], wlds[NW + (k0 + 1) * 64 + col]};
        c = __builtin_amdgcn_wmma_f32_16x16x4_f32(false, aX[kk], false, bR,
                                                  (short)0, c, false, false);
      }
    }
    float bb = bias ? bias[col] : 0.0f;
#pragma unroll
    for (int j = 0; j < 8; ++j) {
      float v = c[j] + bb;
      if (relu) v = fmaxf(v, 0.0f);
      out[(long)(row_base + half * 8 + j) * 64 + col] = v;  // D layout: M=j+8*half
    }
  }
}

// ---------------- small helpers ----------------
__global__ void small_matvec(const float* __restrict__ Wm, const float* __restrict__ v,
                             float* out, int R, int C) {
  int i = blockIdx.x * blockDim.x + threadIdx.x;
  if (i >= R) return;
  float s = 0.f;
  for (int j = 0; j < C; ++j) s += Wm[(long)i * C + j] * v[j];
  out[i] = s;
}

__global__ void rowdot64(const float* __restrict__ X, const float* __restrict__ v,
                         float* out, int N) {
  int n = blockIdx.x * blockDim.x + threadIdx.x;
  if (n >= N) return;
  const float* r = X + (long)n * 64;
  float s = 0.f;
#pragma unroll
  for (int k = 0; k < 64; ++k) s += r[k] * v[k];
  out[n] = s;
}

// ordered-uint float max trick (monotone map)
__device__ inline unsigned f2ord(float f) {
  unsigned u = __float_as_uint(f);
  return (u & 0x80000000u) ? ~u : (u | 0x80000000u);
}
__device__ inline float ord2f(unsigned u) {
  return __uint_as_float((u & 0x80000000u) ? (u ^ 0x80000000u) : ~u);
}

// ---------------- GAT passes ----------------
__global__ void gat_logits(const int* __restrict__ src, const int* __restrict__ dst,
                           const float* __restrict__ ea, const float* __restrict__ wea,
                           const float* __restrict__ ss, const float* __restrict__ dd,
                           float* elog, unsigned* mord, long E) {
  long e = (long)blockIdx.x * blockDim.x + threadIdx.x;
  if (e >= E) return;
  int s = src[e], d = dst[e];
  const float* er = ea + e * 8;
  float t = ss[s] + dd[d];
#pragma unroll
  for (int j = 0; j < 8; ++j) t += er[j] * wea[j];
  t = (t > 0.f) ? t : 0.2f * t;        // leaky_relu(0.2)
  elog[e] = t;
  atomicMax(&mord[d], f2ord(t));
}

__global__ void gat_expz(const int* __restrict__ dst, float* elog,
                         const unsigned* __restrict__ mord, float* z, long E) {
  long e = (long)blockIdx.x * blockDim.x + threadIdx.x;
  if (e >= E) return;
  int d = dst[e];
  float ee = expf(elog[e] - ord2f(mord[d]));
  elog[e] = ee;
  atomicAdd(&z[d], ee);
}

__global__ void gat_accum(const int* __restrict__ src, const int* __restrict__ dst,
                          const float* __restrict__ elog, const float* __restrict__ z,
                          const float* __restrict__ hs, float* out, long E) {
  long t = (long)blockIdx.x * blockDim.x + threadIdx.x;
  long e = t >> 6;
  if (e >= E) return;
  int k = (int)(t & 63);
  int s = src[e], d = dst[e];
  float alpha = elog[e] / (z[d] + 1e-16f);
  atomicAdd(&out[(long)d * 64 + k], alpha * hs[(long)s * 64 + k]);
}

__global__ void bias_relu64(float* x, const float* __restrict__ b, long n) {
  long i = (long)blockIdx.x * blockDim.x + threadIdx.x;
  if (i >= n) return;
  x[i] = fmaxf(x[i] + b[i & 63], 0.f);
}

__global__ void final_dot(const float* __restrict__ X, const float* __restrict__ w,
                          const float* __restrict__ b, float* out, int N) {
  int n = blockIdx.x * blockDim.x + threadIdx.x;
  if (n >= N) return;
  const float* r = X + (long)n * 64;
  float s = b[0];
#pragma unroll
  for (int k = 0; k < 64; ++k) s += r[k] * w[k];
  out[n] = s;
}

// ---------------- host orchestration ----------------
extern "C" void kernel_launch(void* const* d_in, const int* in_sizes, int n_in,
                              void* d_out, int out_size, void* d_ws, size_t ws_size,
                              hipStream_t stream) {
  const float* gx      = (const float*)d_in[0];
  const float* sx      = (const float*)d_in[1];
  const float* ea_hist = (const float*)d_in[2];
  const float* gv0_wl = (const float*)d_in[3],  *gv0_wr = (const float*)d_in[4],  *gv0_b = (const float*)d_in[5];
  const float* gv1_wl = (const float*)d_in[6],  *gv1_wr = (const float*)d_in[7],  *gv1_b = (const float*)d_in[8];
  const float* sv0_wl = (const float*)d_in[9],  *sv0_wr = (const float*)d_in[10], *sv0_b = (const float*)d_in[11];
  const float* sv1_wl = (const float*)d_in[12], *sv1_wr = (const float*)d_in[13], *sv1_b = (const float*)d_in[14];
  const float* gat_ws = (const float*)d_in[15], *gat_wd = (const float*)d_in[16], *gat_we = (const float*)d_in[17];
  const float* gat_as = (const float*)d_in[18], *gat_ad = (const float*)d_in[19], *gat_ae = (const float*)d_in[20];
  const float* gat_b  = (const float*)d_in[21];
  const float* in_wl  = (const float*)d_in[22], *in_wr  = (const float*)d_in[23], *in_b  = (const float*)d_in[24];
  // d_in[25..30]: sv20/sv21 — dead code in the reference, skipped entirely.
  const float* mlp_w  = (const float*)d_in[31], *mlp_b  = (const float*)d_in[32];
  const int* ei_gg   = (const int*)d_in[33];
  const int* ei_ss   = (const int*)d_in[34];
  const int* ei_hist = (const int*)d_in[35];
  const int* ei_in   = (const int*)d_in[36];

  const int NG = in_sizes[0] / 32;
  const int NS = in_sizes[1] / 32;
  const long E_GG = in_sizes[33] / 2;
  const long E_SS = in_sizes[34] / 2;
  const long E_H  = in_sizes[35] / 2;
  const long E_IN = in_sizes[36] / 2;
  const int Nmax = (NG > NS) ? NG : NS;

  // workspace layout (floats)
  float* W = (float*)d_ws;
  const long SLOT = (long)Nmax * 64;
  float* S0 = W;              // aggregation accumulator
  float* S1 = W + SLOT;       // g0 -> s0 -> hs -> in_x
  float* S2 = W + 2 * SLOT;   // g
  float* S3 = W + 3 * SLOT;   // s -> GAT out -> h
  float* V  = W + 4 * SLOT;
  float* cnt  = V;
  float* ssb  = V + Nmax;
  float* dd   = V + 2 * (long)Nmax;
  float* z    = V + 3 * (long)Nmax;
  unsigned* mord = (unsigned*)(V + 4 * (long)Nmax);
  float* elog = V + 5 * (long)Nmax;
  float* wea  = elog + E_H;
  float* wda  = wea + 8;

  const int T = 256;
  auto NB = [](long n) { return (unsigned)((n + 255) / 256); };
  const unsigned LBG = (unsigned)((NG + 127) / 128);  // linear_wmma grid (8 waves * 16 rows)
  const unsigned LBS = (unsigned)((NS + 127) / 128);

  // ---- g = relu(SAGE(relu(SAGE(gx)), ei_gg)) ----
  fill_f32<<<NB((long)NG * 32), T, 0, stream>>>(S0, 0.f, (long)NG * 32);
  fill_f32<<<NB(NG), T, 0, stream>>>(cnt, 0.f, NG);
  sage_agg<<<NB(E_GG * 32), T, 0, stream>>>(ei_gg, ei_gg + E_GG, gx, S0, cnt, E_GG, 5);
  linear_wmma<32, true><<<LBG, 256, 0, stream>>>(S0, cnt, gx, gv0_wl, gv0_wr, gv0_b, S1, NG, 1);

  fill_f32<<<NB((long)NG * 64), T, 0, stream>>>(S0, 0.f, (long)NG * 64);
  fill_f32<<<NB(NG), T, 0, stream>>>(cnt, 0.f, NG);
  sage_agg<<<NB(E_GG * 64), T, 0, stream>>>(ei_gg, ei_gg + E_GG, S1, S0, cnt, E_GG, 6);
  linear_wmma<64, true><<<LBG, 256, 0, stream>>>(S0, cnt, S1, gv1_wl, gv1_wr, gv1_b, S2, NG, 1);

  // ---- s = relu(SAGE(relu(SAGE(sx)), ei_ss)) ----
  fill_f32<<<NB((long)NS * 32), T, 0, stream>>>(S0, 0.f, (long)NS * 32);
  fill_f32<<<NB(NS), T, 0, stream>>>(cnt, 0.f, NS);
  sage_agg<<<NB(E_SS * 32), T, 0, stream>>>(ei_ss, ei_ss + E_SS, sx, S0, cnt, E_SS, 5);
  linear_wmma<32, true><<<LBS, 256, 0, stream>>>(S0, cnt, sx, sv0_wl, sv0_wr, sv0_b, S1, NS, 1);

  fill_f32<<<NB((long)NS * 64), T, 0, stream>>>(S0, 0.f, (long)NS * 64);
  fill_f32<<<NB(NS), T, 0, stream>>>(cnt, 0.f, NS);
  sage_agg<<<NB(E_SS * 64), T, 0, stream>>>(ei_ss, ei_ss + E_SS, S1, S0, cnt, E_SS, 6);
  linear_wmma<64, true><<<LBS, 256, 0, stream>>>(S0, cnt, S1, sv1_wl, sv1_wr, sv1_b, S3, NS, 1);

  // ---- GAT precomputations: wda = wd@a_d, wea = we@a_e, dd = s@wda ----
  small_matvec<<<1, 64, 0, stream>>>(gat_wd, gat_ad, wda, 64, 64);
  small_matvec<<<1, 64, 0, stream>>>(gat_we, gat_ae, wea, 8, 64);
  rowdot64<<<NB(NS), T, 0, stream>>>(S3, wda, dd, NS);

  // hs = g @ ws (plain GEMM via WMMA), ss = hs @ a_s
  linear_wmma<64, false><<<LBG, 256, 0, stream>>>(S2, nullptr, nullptr, gat_ws, nullptr, nullptr, S1, NG, 0);
  rowdot64<<<NB(NG), T, 0, stream>>>(S1, gat_as, ssb, NG);

  // ---- aggregate g over ei_in (S0 holds the running sum until the final linear) ----
  fill_f32<<<NB((long)NS * 64), T, 0, stream>>>(S0, 0.f, (long)NS * 64);
  fill_f32<<<NB(NS), T, 0, stream>>>(cnt, 0.f, NS);
  sage_agg<<<NB(E_IN * 64), T, 0, stream>>>(ei_in, ei_in + E_IN, S2, S0, cnt, E_IN, 6);

  // ---- GAT softmax-attention over ei_hist ----
  fill_u32<<<NB(NS), T, 0, stream>>>(mord, 0x007FFFFFu, NS);  // f2ord(-inf)
  fill_f32<<<NB(NS), T, 0, stream>>>(z, 0.f, NS);
  gat_logits<<<NB(E_H), T, 0, stream>>>(ei_hist, ei_hist + E_H, ea_hist, wea, ssb, dd, elog, mord, E_H);
  gat_expz<<<NB(E_H), T, 0, stream>>>(ei_hist + E_H, elog, mord, z, E_H);
  fill_f32<<<NB((long)NS * 64), T, 0, stream>>>(S3, 0.f, (long)NS * 64);  // s is dead now
  gat_accum<<<NB(E_H * 64), T, 0, stream>>>(ei_hist, ei_hist + E_H, elog, z, S1, S3, E_H);
  bias_relu64<<<NB((long)NS * 64), T, 0, stream>>>(S3, gat_b, (long)NS * 64);  // h in S3

  // ---- in_x = relu(SAGE(mean_in, h)) ----  (hs in S1 is dead)
  linear_wmma<64, true><<<LBS, 256, 0, stream>>>(S0, cnt, S3, in_wl, in_wr, in_b, S1, NS, 1);

  // ---- out = in_x @ mlp_w + mlp_b ----
  final_dot<<<NB(NS), T, 0, stream>>>(S1, mlp_w, mlp_b, (float*)d_out, NS);
}